// Encoder_11879879540958
// MI455X (gfx1250) — compile-verified
//
#include <hip/hip_runtime.h>

#define VOCAB 32000
#define EMB   256
#define HID   1024
#define TLEN  256
#define BATCH 128
#define NG    4096   // 4*HID

#define KC     16                 // K rows staged per LDS stage
#define NPR    (KC / 2)           // 8 pair-rows per stage
#define PST    160                // pair-row stride in dwords (>=128, %64==32)
#define NSTG_W (EMB / KC)         // 16 stages over W
#define NSTG   ((EMB + HID) / KC) // 80 stages total

typedef float v2f __attribute__((ext_vector_type(2)));
typedef float v8f __attribute__((ext_vector_type(8)));

__device__ __forceinline__ float sigf(float x) {
    return 1.0f / (1.0f + expf(-x));
}

__global__ void lstm_init_state(const float* __restrict__ sh,
                                const float* __restrict__ sc,
                                float* __restrict__ h0,
                                float* __restrict__ c0) {
    int i = blockIdx.x * blockDim.x + threadIdx.x;
    if (i < BATCH * HID) { h0[i] = sh[i]; c0[i] = sc[i]; }
}

__global__ void lstm_finalize(const float* __restrict__ h,
                              const float* __restrict__ c,
                              float* __restrict__ hf,
                              float* __restrict__ cf) {
    int i = blockIdx.x * blockDim.x + threadIdx.x;
    if (i < BATCH * HID) { hf[i] = h[i]; cf[i] = c[i]; }
}

// One LSTM timestep: z = [x_t ; h_in] @ [W ; U] + b, gates, masked update.
// Grid: 64 blocks (16 hidden cols -> 64 z-cols incl. gates) x 256 threads (8 waves).
// Weight slice staged through LDS in fragment-ready (k,k+1)-pair layout so each
// WMMA B-fragment is a single conflict-free ds_load_b64.
__global__ __launch_bounds__(256)
void lstm_step(const int*   __restrict__ tokens,   // (B, T)
               const float* __restrict__ emb,      // (VOCAB, EMB)
               const float* __restrict__ W,        // (EMB, 4H)
               const float* __restrict__ U,        // (HID, 4H)
               const float* __restrict__ bias,     // (4H)
               const float* __restrict__ h_in,     // (B, HID)
               float*       __restrict__ h_out,    // (B, HID)
               float*       __restrict__ cbuf,     // (B, HID) in-place
               float*       __restrict__ outs,     // (B, T, HID)
               int t) {
    __shared__ float smem[2][NPR * PST];   // 2 x 5.0 KB

    const int tid  = threadIdx.x;
    const int lane = tid & 31;
    const int wv   = tid >> 5;     // 0..7 -> M-block
    const int hi   = lane >> 4;    // 0/1: K half (A/B frags), M half (C/D)
    const int lrow = lane & 15;
    const int j0   = blockIdx.x * 16;

    // Cooperative staging: 512 vertical pairs per stage, 2 per thread.
    const int c2 = tid & 63;                             // slice col 0..63
    const int pq = tid >> 6;                             // 0..3 -> pair-rows pq, pq+4
    const int gcol = ((c2 >> 4) << 10) + j0 + (c2 & 15); // global z-column

    // A-fragment row (batch index) for this lane.
    const int arow = wv * 16 + lrow;
    const int tokA = tokens[arow * TLEN + t];
    const float* aE = emb  + (size_t)tokA * EMB + 2 * hi;   // K in [0,256)
    const float* aH = h_in + (size_t)arow * HID + 2 * hi;   // K in [256,1280)

    v8f acc[4];
    #pragma unroll
    for (int g = 0; g < 4; ++g) acc[g] = (v8f){0,0,0,0,0,0,0,0};

    // ---- prologue: prefetch + commit stage 0 (W rows 0..15) ----
    v2f pre[2];
    {
        #pragma unroll
        for (int i = 0; i < 2; ++i) {
            const float* p = W + (size_t)(2 * (pq + 4 * i)) * NG + gcol;
            pre[i].x = p[0];
            pre[i].y = p[NG];
        }
        #pragma unroll
        for (int i = 0; i < 2; ++i)
            *(v2f*)&smem[0][(pq + 4 * i) * PST + c2 * 2] = pre[i];
    }
    __syncthreads();

    for (int s = 0; s < NSTG; ++s) {
        // Prefetch next stage's weight slice into registers (hidden behind WMMAs).
        const bool have_next = (s + 1 < NSTG);
        if (have_next) {
            const int sn = s + 1;
            const float* src = (sn < NSTG_W)
                ? (W + (size_t)sn * KC * NG)
                : (U + (size_t)(sn - NSTG_W) * KC * NG);
            #pragma unroll
            for (int i = 0; i < 2; ++i) {
                const float* p = src + (size_t)(2 * (pq + 4 * i)) * NG + gcol;
                pre[i].x = p[0];
                pre[i].y = p[NG];
            }
        }

        // Compute current stage: 4 K-chunks of 4; B frags = single ds_load_b64.
        const float* aptr = (s < NSTG_W) ? (aE + s * KC)
                                         : (aH + (s - NSTG_W) * KC);
        const float* bb = &smem[s & 1][hi * PST + lrow * 2];
        #pragma unroll
        for (int kc = 0; kc < KC; kc += 4) {
            v2f a = *(const v2f*)(aptr + kc);
            #pragma unroll
            for (int g = 0; g < 4; ++g) {
                v2f bf = *(const v2f*)(bb + (kc >> 1) * PST + g * 32);
                acc[g] = __builtin_amdgcn_wmma_f32_16x16x4_f32(
                    false, a, false, bf, (short)0, acc[g], false, false);
            }
        }

        // Commit prefetched slice to the other LDS buffer, then sync.
        if (have_next) {
            #pragma unroll
            for (int i = 0; i < 2; ++i)
                *(v2f*)&smem[(s + 1) & 1][(pq + 4 * i) * PST + c2 * 2] = pre[i];
        }
        __syncthreads();
    }

    // ---- Epilogue: bias + gates + masked state update ----
    float bv[4];
    #pragma unroll
    for (int g = 0; g < 4; ++g) bv[g] = bias[(g << 10) + j0 + lrow];

    const int j = j0 + lrow;
    #pragma unroll
    for (int r = 0; r < 8; ++r) {
        const int row = wv * 16 + hi * 8 + r;   // C/D layout: lanes16-31 -> M+8
        const bool m  = tokens[row * TLEN + t] != 0;

        float zi = acc[0][r] + bv[0];
        float zf = acc[1][r] + bv[1];
        float zg = acc[2][r] + bv[2];
        float zo = acc[3][r] + bv[3];

        const size_t idx = (size_t)row * HID + j;
        float c_old = cbuf[idx];
        float c_new = sigf(zf) * c_old + sigf(zi) * tanhf(zg);
        float h_new = sigf(zo) * tanhf(c_new);
        float h_prev = h_in[idx];

        float h_sel = m ? h_new : h_prev;
        float c_sel = m ? c_new : c_old;

        cbuf[idx]  = c_sel;
        h_out[idx] = h_sel;
        outs[((size_t)row * TLEN + t) * HID + j] = h_sel;
    }
}

extern "C" void kernel_launch(void* const* d_in, const int* in_sizes, int n_in,
                              void* d_out, int out_size, void* d_ws, size_t ws_size,
                              hipStream_t stream) {
    const int*   tokens = (const int*)  d_in[0];  // (B,T)
    const float* sh     = (const float*)d_in[1];  // (B,HID)
    const float* sc     = (const float*)d_in[2];  // (B,HID)
    const float* emb    = (const float*)d_in[3];  // (VOCAB,EMB)
    const float* W      = (const float*)d_in[4];  // (EMB,4H)
    const float* U      = (const float*)d_in[5];  // (HID,4H)
    const float* bias   = (const float*)d_in[6];  // (4H)

    float* out  = (float*)d_out;
    float* outs = out;                                   // (B,T,HID)
    float* hf   = out + (size_t)BATCH * TLEN * HID;      // (B,HID)
    float* cf   = hf + (size_t)BATCH * HID;              // (B,HID)

    float* h0 = (float*)d_ws;
    float* h1 = h0 + (size_t)BATCH * HID;
    float* c0 = h1 + (size_t)BATCH * HID;

    const int nstate = BATCH * HID;
    lstm_init_state<<<(nstate + 255) / 256, 256, 0, stream>>>(sh, sc, h0, c0);

    float* hin  = h0;
    float* hout = h1;
    for (int t = 0; t < TLEN; ++t) {
        lstm_step<<<HID / 16, 256, 0, stream>>>(
            tokens, emb, W, U, bias, hin, hout, c0, outs, t);
        float* tmp = hin; hin = hout; hout = tmp;
    }
    // After an even number of steps + swaps, final h is in `hin`.
    lstm_finalize<<<(nstate + 255) / 256, 256, 0, stream>>>(hin, c0, hf, cf);
}